// QuantumReuploadPQC_41283225649815
// MI455X (gfx1250) — compile-verified
//
#include <hip/hip_runtime.h>
#include <hip/hip_bf16.h>
#include <math.h>

#define N_WIRES   8
#define NUM_BLOCK 3
#define BATCH     64
#define DIM       256
#define RHO_ELEMS (DIM * DIM)      // 65536 complex64 per batch element
#define N_STEPS   14               // 12 pair steps + 2 (CZ + trailing-noise) steps
#define P_BITFLIP 0.02f
#define P_DEPOL   0.03f

typedef __attribute__((ext_vector_type(2))) float v2f;
typedef __attribute__((ext_vector_type(8))) float v8f;

// ---------------- complex helpers ----------------
__device__ __forceinline__ float2 cmk(float re, float im) { float2 r; r.x = re; r.y = im; return r; }
__device__ __forceinline__ float2 cmul(float2 a, float2 b) {
    return cmk(a.x * b.x - a.y * b.y, a.x * b.y + a.y * b.x);
}
__device__ __forceinline__ float2 cmulc(float2 a, float2 b) {  // a * conj(b)
    return cmk(a.x * b.x + a.y * b.y, a.y * b.x - a.x * b.y);
}
__device__ __forceinline__ float2 cadd(float2 a, float2 b) { return cmk(a.x + b.x, a.y + b.y); }

// M[(a,b),(c,d)] = sum_k K_k[a][c] * conj(K_k[b][d])
__device__ void superop_from_kraus(float2 M[4][4], const float2 K[][2][2], int nk) {
    for (int r = 0; r < 4; ++r)
        for (int c = 0; c < 4; ++c) M[r][c] = cmk(0.f, 0.f);
    for (int k = 0; k < nk; ++k)
        for (int a = 0; a < 2; ++a)
            for (int b = 0; b < 2; ++b)
                for (int c = 0; c < 2; ++c)
                    for (int d = 0; d < 2; ++d)
                        M[a * 2 + b][c * 2 + d] =
                            cadd(M[a * 2 + b][c * 2 + d], cmulc(K[k][a][c], K[k][b][d]));
}

__device__ void matmul4(float2 C[4][4], const float2 A[4][4], const float2 B[4][4]) {
    for (int r = 0; r < 4; ++r)
        for (int c = 0; c < 4; ++c) {
            float2 acc = cmk(0.f, 0.f);
            for (int k = 0; k < 4; ++k) acc = cadd(acc, cmul(A[r][k], B[k][c]));
            C[r][c] = acc;
        }
}

// PennyLane Rot(phi,theta,omega) = RZ(omega) RY(theta) RZ(phi)
__device__ void rot_u(float2 U[2][2], float phi, float theta, float omega) {
    float c  = cosf(theta * 0.5f);
    float sn = sinf(theta * 0.5f);
    float ap = 0.5f * (phi + omega);
    float am = 0.5f * (phi - omega);
    float2 ep = cmk(cosf(ap), -sinf(ap));   // e^{-i(phi+omega)/2}
    float2 em = cmk(cosf(am), -sinf(am));   // e^{-i(phi-omega)/2}
    U[0][0] = cmk(ep.x * c, ep.y * c);
    U[0][1] = cmk(-em.x * sn, em.y * sn);   // -conj(em)*s
    U[1][0] = cmk(em.x * sn, em.y * sn);
    U[1][1] = cmk(ep.x * c, -ep.y * c);     // conj(ep)*c
}

__device__ void superop_unitary(float2 M[4][4], const float2 U[2][2]) {
    float2 K[1][2][2];
    K[0][0][0] = U[0][0]; K[0][0][1] = U[0][1];
    K[0][1][0] = U[1][0]; K[0][1][1] = U[1][1];
    superop_from_kraus(M, K, 1);
}

__device__ void superop_bitflip(float2 M[4][4]) {
    float a = sqrtf(1.f - P_BITFLIP), b = sqrtf(P_BITFLIP);
    float2 K[2][2][2];
    K[0][0][0] = cmk(a, 0); K[0][0][1] = cmk(0, 0); K[0][1][0] = cmk(0, 0); K[0][1][1] = cmk(a, 0);
    K[1][0][0] = cmk(0, 0); K[1][0][1] = cmk(b, 0); K[1][1][0] = cmk(b, 0); K[1][1][1] = cmk(0, 0);
    superop_from_kraus(M, K, 2);
}

__device__ void superop_depol(float2 M[4][4]) {
    float a = sqrtf(1.f - P_DEPOL), b = sqrtf(P_DEPOL / 3.f);
    float2 K[4][2][2];
    K[0][0][0] = cmk(a, 0);  K[0][0][1] = cmk(0, 0);  K[0][1][0] = cmk(0, 0);  K[0][1][1] = cmk(a, 0);
    K[1][0][0] = cmk(0, 0);  K[1][0][1] = cmk(b, 0);  K[1][1][0] = cmk(b, 0);  K[1][1][1] = cmk(0, 0);
    K[2][0][0] = cmk(0, 0);  K[2][0][1] = cmk(0, -b); K[2][1][0] = cmk(0, b);  K[2][1][1] = cmk(0, 0);
    K[3][0][0] = cmk(b, 0);  K[3][0][1] = cmk(0, 0);  K[3][1][0] = cmk(0, 0);  K[3][1][1] = cmk(-b, 0);
    superop_from_kraus(M, K, 4);
}

__device__ void superop_ampdamp(float2 M[4][4], float g) {
    float2 K[2][2][2];
    K[0][0][0] = cmk(1, 0); K[0][0][1] = cmk(0, 0);
    K[0][1][0] = cmk(0, 0); K[0][1][1] = cmk(sqrtf(fmaxf(0.f, 1.f - g)), 0);
    K[1][0][0] = cmk(0, 0); K[1][0][1] = cmk(sqrtf(fmaxf(0.f, g)), 0);
    K[1][1][0] = cmk(0, 0); K[1][1][1] = cmk(0, 0);
    superop_from_kraus(M, K, 2);
}

// full single-wire step: AD(sig(x0)) . DP . BF . Rot(w) . Rot(x)
__device__ void wire_step_superop(float2 M[4][4], const float* x3, const float* w3) {
    float2 U1[2][2], U2[2][2];
    rot_u(U1, x3[0], x3[1], x3[2]);
    rot_u(U2, w3[0], w3[1], w3[2]);
    float2 M1[4][4], M2[4][4], M3[4][4], T[4][4];
    superop_unitary(M1, U1);
    superop_unitary(M2, U2);
    matmul4(T, M2, M1);
    superop_bitflip(M3);
    matmul4(M1, M3, T);
    superop_depol(M2);
    matmul4(T, M2, M1);
    float g = 1.f / (1.f + expf(-x3[0]));
    superop_ampdamp(M3, g);
    matmul4(M, M3, T);
}

__device__ void noise_superop(float2 M[4][4]) {   // depol . bitflip
    float2 Bf[4][4], Dp[4][4];
    superop_bitflip(Bf);
    superop_depol(Dp);
    matmul4(M, Dp, Bf);
}

__device__ void identity_superop(float2 M[4][4]) {
    for (int r = 0; r < 4; ++r)
        for (int c = 0; c < 4; ++c) M[r][c] = cmk(r == c ? 1.f : 0.f, 0.f);
}

// ------------- kernel A: per-(batch, step) 2-wire pair superoperator, realified 32x32 -------------
// step map: 0-3 blk0 pairs, 4 = CZ+noise, 5-8 blk1 pairs, 9 = CZ+noise, 10-13 blk2 pairs
__global__ void build_pair_superops(const float* __restrict__ x, const float* __restrict__ wgt,
                                    float* __restrict__ Rtab) {
    int tid = blockIdx.x * blockDim.x + threadIdx.x;
    if (tid >= BATCH * N_STEPS) return;
    int b = tid / N_STEPS;
    int s = tid % N_STEPS;

    float2 Se[4][4], So[4][4];   // even wire (2p) acts on (c1,d1); odd wire (2p+1) on (c0,d0)
    if (s == 4 || s == 9) {
        identity_superop(Se);    // wire 6 untouched
        noise_superop(So);       // trailing noise on wire 7
    } else {
        int t = s - (s > 4 ? 1 : 0) - (s > 9 ? 1 : 0);
        int blk = t >> 2, p = t & 3;
        int we = 2 * p, wo = 2 * p + 1;
        wire_step_superop(Se, x + b * (3 * N_WIRES) + 3 * we, wgt + (blk * N_WIRES + we) * 3);
        wire_step_superop(So, x + b * (3 * N_WIRES) + 3 * wo, wgt + (blk * N_WIRES + wo) * 3);
    }
    // M16[v][v'] = Se[2c1+d1][2c1'+d1'] * So[2c0+d0][2c0'+d0'],  v = (c1 c0 d1 d0)
    // realified 32x32 with interleaved [Re,Im] comps
    float* R = Rtab + (size_t)(b * N_STEPS + s) * 1024;
    for (int v = 0; v < 16; ++v) {
        int re_ = 2 * ((v >> 3) & 1) + ((v >> 1) & 1);
        int ro_ = 2 * ((v >> 2) & 1) + (v & 1);
        for (int vp = 0; vp < 16; ++vp) {
            int rep = 2 * ((vp >> 3) & 1) + ((vp >> 1) & 1);
            int rop = 2 * ((vp >> 2) & 1) + (vp & 1);
            float2 val = cmul(Se[re_][rep], So[ro_][rop]);
            R[(2 * v) * 32 + 2 * vp]         = val.x;
            R[(2 * v) * 32 + 2 * vp + 1]     = -val.y;
            R[(2 * v + 1) * 32 + 2 * vp]     = val.y;
            R[(2 * v + 1) * 32 + 2 * vp + 1] = val.x;
        }
    }
}

// ---------------- init rho = |0..0><0..0| ----------------
__global__ void init_rho(float2* __restrict__ rho) {
    unsigned tid = blockIdx.x * 256u + threadIdx.x;
    rho[tid] = cmk((tid & 65535u) == 0u ? 1.f : 0.f, 0.f);
}

// -------- WMMA pair sweep: apply 32x32 real superop to 4096 16-complex blocks / batch --------
// pair (2p,2p+1): bra bits at {ql, ql+1}, ket bits at {ql+8, ql+9}, ql = 6-2p.
// If doCZ, multiply each input entry by the CZ-chain diagonal sign (CZ precedes the noise step).
__global__ __launch_bounds__(256) void sweep_pair(float2* __restrict__ rho,
                                                  const float* __restrict__ Rtab,
                                                  int step, int ql, int doCZ) {
    int lane = threadIdx.x & 31;
    int W    = blockIdx.x * 8 + (threadIdx.x >> 5);   // 4096 waves total
    int b    = W >> 6;                                 // 64 waves per batch element
    int chunk = W & 63;                                // 64 groups per wave (4 iters x 16)
    int m  = lane & 15;
    int hi = lane >> 4;

    // A tiles: A_lo rows = R[0..15], A_hi rows = R[16..31]; 8 K=4 chunks each.
    const float* R = Rtab + (size_t)(b * N_STEPS + step) * 1024;
    v2f alo[8], ahi[8];
#pragma unroll
    for (int k = 0; k < 8; ++k) {
        int col = 4 * k + 2 * hi;
        alo[k][0] = R[m * 32 + col];          alo[k][1] = R[m * 32 + col + 1];
        ahi[k][0] = R[(m + 16) * 32 + col];   ahi[k][1] = R[(m + 16) * 32 + col + 1];
    }

    float2* rb = rho + ((size_t)b << 16);
    unsigned maskLow = (1u << ql) - 1u;

    for (int it = 0; it < 4; ++it) {
        unsigned g    = ((unsigned)chunk << 6) + ((unsigned)it << 4) + (unsigned)m;  // 12-bit group
        unsigned low  = g & maskLow;
        unsigned mid  = (g >> ql) & 63u;
        unsigned high = g >> (ql + 6);
        unsigned f = low | (mid << (ql + 2)) | (high << (ql + 10)); // zeros at ql,ql+1,ql+8,ql+9

        // B: chunk k covers in-comps 4k..4k+3; lane half hi carries entry v = 2k+hi as float2
        v2f bv[8];
#pragma unroll
        for (int k = 0; k < 8; ++k) {
            unsigned v   = 2u * (unsigned)k + (unsigned)hi;
            unsigned off = ((v >> 2) << (ql + 8)) | ((v & 3u) << ql);
            float2 e = rb[f + off];
            if (doCZ) {
                unsigned fe = f + off;
                unsigned kk = fe >> 8, br = fe & 255u;
                int par = (__popc(kk & (kk >> 1)) + __popc(br & (br >> 1))) & 1;
                float sgn = 1.f - 2.f * (float)par;
                e.x *= sgn; e.y *= sgn;
            }
            bv[k][0] = e.x; bv[k][1] = e.y;
        }

        v8f dlo = {}, dhi = {};
#pragma unroll
        for (int k = 0; k < 8; ++k) {
            dlo = __builtin_amdgcn_wmma_f32_16x16x4_f32(false, alo[k], false, bv[k], (short)0, dlo, false, false);
            dhi = __builtin_amdgcn_wmma_f32_16x16x4_f32(false, ahi[k], false, bv[k], (short)0, dhi, false, false);
        }

        // D_lo: lane holds comps hi*8+0..7 = entries 4hi..4hi+3; D_hi: entries 8+4hi..8+4hi+3
#pragma unroll
        for (int j = 0; j < 4; ++j) {
            unsigned v1  = 4u * (unsigned)hi + (unsigned)j;
            unsigned o1  = ((v1 >> 2) << (ql + 8)) | ((v1 & 3u) << ql);
            rb[f + o1] = cmk(dlo[2 * j], dlo[2 * j + 1]);
            unsigned v2  = v1 + 8u;
            unsigned o2  = ((v2 >> 2) << (ql + 8)) | ((v2 & 3u) << ql);
            rb[f + o2] = cmk(dhi[2 * j], dhi[2 * j + 1]);
        }
    }
}

// ---------------- diagonal -> PauliZ expectation values ----------------
__global__ void expect_kernel(const float2* __restrict__ rho, float* __restrict__ out) {
    __shared__ float sp[DIM];
    int b = blockIdx.x, t = threadIdx.x;
    sp[t] = rho[((size_t)b << 16) + (size_t)t * 257].x;   // diag element k*256+k
    __syncthreads();
    int w = t >> 5, lane = t & 31;
    float s = 0.f;
    for (int j = 0; j < 8; ++j) {
        int k = lane + 32 * j;
        float sgn = ((k >> (7 - w)) & 1) ? -1.f : 1.f;    // wire w = bit (7-w) of k
        s += sp[k] * sgn;
    }
    for (int off = 16; off; off >>= 1) s += __shfl_xor(s, off, 32);
    if (lane == 0) out[b * N_WIRES + w] = s;
}

extern "C" void kernel_launch(void* const* d_in, const int* in_sizes, int n_in,
                              void* d_out, int out_size, void* d_ws, size_t ws_size,
                              hipStream_t stream) {
    const float* x   = (const float*)d_in[0];   // [64, 24]
    const float* wgt = (const float*)d_in[1];   // [3, 8, 3]
    float* out = (float*)d_out;                 // [64, 8]

    float* Rtab = (float*)d_ws;
    size_t rtab_bytes = ((size_t)BATCH * N_STEPS * 1024 * sizeof(float) + 255) & ~(size_t)255;
    float2* rho = (float2*)((char*)d_ws + rtab_bytes);  // 64 x 65536 complex64 = 32 MiB

    build_pair_superops<<<(BATCH * N_STEPS + 255) / 256, 256, 0, stream>>>(x, wgt, Rtab);
    init_rho<<<(BATCH * RHO_ELEMS) / 256, 256, 0, stream>>>(rho);

    int step = 0;
    for (int blk = 0; blk < NUM_BLOCK; ++blk) {
        for (int p = 0; p < 4; ++p) {
            sweep_pair<<<512, 256, 0, stream>>>(rho, Rtab, step, 6 - 2 * p, 0);
            ++step;
        }
        if (blk < NUM_BLOCK - 1) {
            // CZ chain (diagonal signs, folded into loads) + trailing noise on wire 7: pair (6,7)
            sweep_pair<<<512, 256, 0, stream>>>(rho, Rtab, step, 0, 1);
            ++step;
        }
    }
    expect_kernel<<<BATCH, 256, 0, stream>>>(rho, out);
}